// TGCNBaseline_34711925686420
// MI455X (gfx1250) — compile-verified
//
#include <hip/hip_runtime.h>
#include <math.h>

// ---------------- problem constants (match reference) ----------------
#define NN     50000      // nodes
#define NE     800000     // edges
#define NC     64         // census features
#define HID    128
#define NL     2
#define NETH   10
#define EDIM   32
#define BATCH  16384
#define KENC   68         // 65 padded up to multiple of 4
#define ZDIM   (HID + EDIM)
#define LN_EPS 1e-5f

#define CEILDIV(a,b) (((a)+(b)-1)/(b))

typedef __attribute__((ext_vector_type(2))) float v2f;
typedef __attribute__((ext_vector_type(8))) float v8f;

__device__ __forceinline__ float warp_sum(float v) {
  #pragma unroll
  for (int o = 16; o > 0; o >>= 1) v += __shfl_xor(v, o, 32);
  return v;
}

// ---------------- WMMA fp32 GEMM: C[M,N] = A[M,K] @ B + bias ----------------
// TRANSB=false: B is row-major [K,N].  TRANSB=true: B is row-major [N,K] (i.e. A@B^T).
// Grid: x = N/16 tiles, y = ceil(M/16 / 4).  Block = 128 threads (4 waves),
// each wave computes one 16x16 tile; the block's K x 16 B-panel is staged in LDS.
template<bool TRANSB>
__global__ __launch_bounds__(128)
void k_gemm(const float* __restrict__ A, const float* __restrict__ B,
            const float* __restrict__ bias, float* __restrict__ C,
            int M, int N, int K) {
  __shared__ float Bs[160 * 16];   // max K = 160
  const int tn = blockIdx.x << 4;

  // cooperative stage of B tile (K rows x 16 cols), row-major in LDS
  for (int i = threadIdx.x; i < K * 16; i += blockDim.x) {
    const int kk = i >> 4, nn = i & 15;
    Bs[i] = TRANSB ? B[(size_t)(tn + nn) * K + kk]
                   : B[(size_t)kk * N + (tn + nn)];
  }
  __syncthreads();

  const int wv = threadIdx.x >> 5;
  const int tm = (blockIdx.y * (blockDim.x >> 5) + wv) << 4;
  if (tm >= M) return;                 // wave-uniform: EXEC stays all-ones below

  const int lane = threadIdx.x & 31;
  const int mn = lane & 15;            // M index within A tile / N index within B tile
  const int kb = (lane >> 4) << 1;     // K sub-offset: lanes 0-15 -> 0, lanes 16-31 -> 2
  const float* Arow = A + (size_t)(tm + mn) * K + kb;
  const float* Bp = Bs + mn;

  v8f acc0 = {0.f,0.f,0.f,0.f,0.f,0.f,0.f,0.f};
  v8f acc1 = {0.f,0.f,0.f,0.f,0.f,0.f,0.f,0.f};
  int k = 0;
  for (; k + 8 <= K; k += 8) {
    { v2f a, b;
      a.x = Arow[k];     a.y = Arow[k + 1];
      b.x = Bp[(k + kb) * 16];     b.y = Bp[(k + kb + 1) * 16];
      acc0 = __builtin_amdgcn_wmma_f32_16x16x4_f32(false, a, false, b, (short)0, acc0, false, false); }
    { v2f a, b;
      a.x = Arow[k + 4]; a.y = Arow[k + 5];
      b.x = Bp[(k + 4 + kb) * 16]; b.y = Bp[(k + 5 + kb) * 16];
      acc1 = __builtin_amdgcn_wmma_f32_16x16x4_f32(false, a, false, b, (short)0, acc1, false, false); }
  }
  for (; k < K; k += 4) {
    v2f a, b;
    a.x = Arow[k]; a.y = Arow[k + 1];
    b.x = Bp[(k + kb) * 16]; b.y = Bp[(k + kb + 1) * 16];
    acc0 = __builtin_amdgcn_wmma_f32_16x16x4_f32(false, a, false, b, (short)0, acc0, false, false);
  }

  // C/D layout: VGPR v, lanes 0-15 -> row tm+v, lanes 16-31 -> row tm+8+v; col = tn + lane%16
  const int row0 = tm + ((lane >> 4) << 3);
  const int col  = tn + mn;
  const float bv = bias ? bias[col] : 0.0f;
  #pragma unroll
  for (int v = 0; v < 8; ++v)
    C[(size_t)(row0 + v) * N + col] = acc0[v] + acc1[v] + bv;
}

// ---------------- graph degree ----------------
__global__ void k_deg_atomic(const int* __restrict__ dst, const float* __restrict__ w,
                             float* __restrict__ deg, int E) {
  int e = blockIdx.x * blockDim.x + threadIdx.x;
  if (e < E) unsafeAtomicAdd(&deg[dst[e]], w[e]);
}

__global__ void k_deg_fin(float* __restrict__ deg, int n) {
  int i = blockIdx.x * blockDim.x + threadIdx.x;
  if (i < n) deg[i] = rsqrtf(deg[i] + 1.0f);
}

// ---------------- feature build: feat = [census | sum(pop)] zero-padded to KENC ----------------
__global__ void k_feat(const float* __restrict__ census, const float* __restrict__ pop,
                       float* __restrict__ feat) {
  int idx = blockIdx.x * blockDim.x + threadIdx.x;
  if (idx >= NN * KENC) return;
  int r = idx / KENC, c = idx % KENC;
  float v = 0.0f;
  if (c < NC) v = census[(size_t)r * NC + c];
  else if (c == NC) {
    #pragma unroll
    for (int j = 0; j < NETH; ++j) v += pop[(size_t)r * NETH + j];
  }
  feat[idx] = v;
}

__global__ void k_padW1(const float* __restrict__ W, float* __restrict__ Wp) {
  int idx = blockIdx.x * blockDim.x + threadIdx.x;
  if (idx >= KENC * HID) return;
  int r = idx / HID, c = idx % HID;
  Wp[idx] = (r < NC + 1) ? W[r * HID + c] : 0.0f;
}

// ---------------- layer-norm (+ optional ReLU), in place, one wave per row ----------------
__global__ void k_ln(float* __restrict__ X, const float* __restrict__ g,
                     const float* __restrict__ b, int M, int R, int relu) {
  int row = blockIdx.x * (blockDim.x >> 5) + (threadIdx.x >> 5);
  if (row >= M) return;
  int lane = threadIdx.x & 31;
  float* xr = X + (size_t)row * R;
  int cnt = R >> 5;                 // R in {64,128} -> 2 or 4
  float vals[4];
  float s = 0.f;
  for (int i = 0; i < cnt; ++i) { vals[i] = xr[lane + (i << 5)]; s += vals[i]; }
  s = warp_sum(s);
  float mean = s / (float)R;
  float var = 0.f;
  for (int i = 0; i < cnt; ++i) { float d = vals[i] - mean; var += d * d; }
  var = warp_sum(var) / (float)R;
  float inv = rsqrtf(var + LN_EPS);
  for (int i = 0; i < cnt; ++i) {
    int c = lane + (i << 5);
    float y = (vals[i] - mean) * inv * g[c] + b[c];
    if (relu) y = fmaxf(y, 0.f);
    xr[c] = y;
  }
}

// ---------------- edge scatter: agg[dst] += xl[src] * (dinv[src]*w*dinv[dst]) ----------------
__global__ void k_edge_scatter(const int* __restrict__ src, const int* __restrict__ dst,
                               const float* __restrict__ w, const float* __restrict__ dinv,
                               const float* __restrict__ xl, float* __restrict__ agg, int E) {
  int e = blockIdx.x * (blockDim.x >> 5) + (threadIdx.x >> 5);
  if (e >= E) return;
  int lane = threadIdx.x & 31;
  int s = src[e], d = dst[e];
  float nrm = dinv[s] * w[e] * dinv[d];
  const float4* xs = (const float4*)(xl + (size_t)s * HID);
  float4 v = xs[lane];               // 32 lanes x float4 = 128 features
  float* ad = agg + (size_t)d * HID + lane * 4;
  unsafeAtomicAdd(ad + 0, v.x * nrm);
  unsafeAtomicAdd(ad + 1, v.y * nrm);
  unsafeAtomicAdd(ad + 2, v.z * nrm);
  unsafeAtomicAdd(ad + 3, v.w * nrm);
}

// ---------------- self-loop term + bias: agg += xl*dinv^2 + b ----------------
__global__ void k_gcn_fin(float* __restrict__ agg, const float* __restrict__ xl,
                          const float* __restrict__ dinv, const float* __restrict__ bias,
                          int n) {
  int idx = blockIdx.x * blockDim.x + threadIdx.x;
  if (idx >= n) return;
  int r = idx / HID, c = idx % HID;
  float di = dinv[r];
  agg[idx] += xl[idx] * di * di + bias[c];
}

// ---------------- GRU gate fusion ----------------
__global__ void k_gru(const float* __restrict__ gi, const float* __restrict__ gh,
                      const float* __restrict__ hprev, float* __restrict__ hnew, int n) {
  int idx = blockIdx.x * blockDim.x + threadIdx.x;
  if (idx >= n) return;
  int r = idx / HID, c = idx % HID;
  const float* gir = gi + (size_t)r * 3 * HID;
  const float* ghr = gh + (size_t)r * 3 * HID;
  float rr = 1.f / (1.f + expf(-(gir[c] + ghr[c])));
  float zz = 1.f / (1.f + expf(-(gir[HID + c] + ghr[HID + c])));
  float nn = tanhf(gir[2 * HID + c] + rr * ghr[2 * HID + c]);
  hnew[idx] = (1.f - zz) * nn + zz * hprev[idx];
}

// ---------------- head gather/concat ----------------
__global__ void k_gather(const float* __restrict__ ctx, const int* __restrict__ nidx,
                         const float* __restrict__ eth, const int* __restrict__ eid,
                         float* __restrict__ z) {
  int idx = blockIdx.x * blockDim.x + threadIdx.x;
  if (idx >= BATCH * ZDIM) return;
  int b = idx / ZDIM, c = idx % ZDIM;
  z[idx] = (c < HID) ? ctx[(size_t)nidx[b] * HID + c]
                     : eth[eid[b] * EDIM + (c - HID)];
}

// ---------------- final dot + softplus, one wave per sample ----------------
__global__ void k_head(const float* __restrict__ h2, const float* __restrict__ W3,
                       const float* __restrict__ b3, float* __restrict__ out) {
  int row = blockIdx.x * (blockDim.x >> 5) + (threadIdx.x >> 5);
  if (row >= BATCH) return;
  int lane = threadIdx.x & 31;
  const float* hr = h2 + (size_t)row * 64;
  float p = hr[lane] * W3[lane] + hr[lane + 32] * W3[lane + 32];
  p = warp_sum(p);
  if (lane == 0) {
    float x = p + b3[0];
    out[row] = fmaxf(x, 0.f) + log1pf(expf(-fabsf(x)));   // stable softplus
  }
}

// ---------------- host side ----------------
static void gemm(const float* A, const float* B, const float* bias, float* C,
                 int M, int N, int K, bool transB, hipStream_t st) {
  dim3 blk(128);
  dim3 grd(N / 16, CEILDIV(M / 16, 4));
  if (transB) k_gemm<true ><<<grd, blk, 0, st>>>(A, B, bias, C, M, N, K);
  else        k_gemm<false><<<grd, blk, 0, st>>>(A, B, bias, C, M, N, K);
}

extern "C" void kernel_launch(void* const* d_in, const int* in_sizes, int n_in,
                              void* d_out, int out_size, void* d_ws, size_t ws_size,
                              hipStream_t stream) {
  (void)in_sizes; (void)n_in; (void)out_size; (void)ws_size;
  // inputs (setup_inputs order)
  const float* census  = (const float*)d_in[0];
  const float* pop     = (const float*)d_in[1];
  const int*   ei      = (const int*)  d_in[2];
  const float* ew      = (const float*)d_in[3];
  const int*   eth_id  = (const int*)  d_in[4];
  const int*   nidx    = (const int*)  d_in[5];
  const float* enc_W1  = (const float*)d_in[6];
  const float* enc_b1  = (const float*)d_in[7];
  const float* enc_g1  = (const float*)d_in[8];
  const float* enc_be1 = (const float*)d_in[9];
  const float* enc_W2  = (const float*)d_in[10];
  const float* enc_b2  = (const float*)d_in[11];
  const float* enc_g2  = (const float*)d_in[12];
  const float* enc_be2 = (const float*)d_in[13];
  const float* gcn_W   = (const float*)d_in[14];
  const float* gcn_b   = (const float*)d_in[15];
  const float* gru_Wih = (const float*)d_in[16];
  const float* gru_Whh = (const float*)d_in[17];
  const float* gru_bih = (const float*)d_in[18];
  const float* gru_bhh = (const float*)d_in[19];
  const float* tg_g    = (const float*)d_in[20];
  const float* tg_b    = (const float*)d_in[21];
  const float* eth_tab = (const float*)d_in[22];
  const float* p_W1    = (const float*)d_in[23];
  const float* p_b1    = (const float*)d_in[24];
  const float* p_g1    = (const float*)d_in[25];
  const float* p_be1   = (const float*)d_in[26];
  const float* p_W2    = (const float*)d_in[27];
  const float* p_b2    = (const float*)d_in[28];
  const float* p_g2    = (const float*)d_in[29];
  const float* p_be2   = (const float*)d_in[30];
  const float* p_W3    = (const float*)d_in[31];
  const float* p_b3    = (const float*)d_in[32];
  float* out = (float*)d_out;

  const int* src = ei;
  const int* dst = ei + NE;

  // workspace carve-up (fp32)
  const size_t NH = (size_t)NN * HID;
  float* p = (float*)d_ws;
  float* dinv  = p; p += NN;                    // degree accumulator -> rsqrt(deg+1)
  float* wpad  = p; p += (size_t)KENC * HID;    // padded enc_W1
  float* feat  = p; p += (size_t)NN * KENC;
  float* x1    = p; p += NH;
  float* x2    = p; p += NH;                    // encoder output (layer-0 input)
  float* oA0   = p; p += NH;
  float* oA1   = p; p += NH;
  float* oB0   = p; p += NH;
  float* oB1   = p; p += NH;
  float* hzero = p; p += NH;
  float* xl    = p; p += NH;
  float* agg   = p; p += NH;
  float* gi    = p; p += (size_t)NN * 3 * HID;
  float* gh    = p; p += (size_t)NN * 3 * HID;
  float* zb    = p; p += (size_t)BATCH * ZDIM;
  float* h1    = p; p += (size_t)BATCH * HID;
  float* h2    = p; p += (size_t)BATCH * (HID / 2);

  // ---- GCN normalization ----
  hipMemsetAsync(dinv, 0, NN * sizeof(float), stream);
  k_deg_atomic<<<CEILDIV(NE, 256), 256, 0, stream>>>(dst, ew, dinv, NE);
  k_deg_fin<<<CEILDIV(NN, 256), 256, 0, stream>>>(dinv, NN);

  // ---- encoder ----
  k_feat<<<CEILDIV(NN * KENC, 256), 256, 0, stream>>>(census, pop, feat);
  k_padW1<<<CEILDIV(KENC * HID, 256), 256, 0, stream>>>(enc_W1, wpad);
  gemm(feat, wpad, enc_b1, x1, NN, HID, KENC, false, stream);
  k_ln<<<CEILDIV(NN, 8), 256, 0, stream>>>(x1, enc_g1, enc_be1, NN, HID, 1);
  gemm(x1, enc_W2, enc_b2, x2, NN, HID, HID, false, stream);
  k_ln<<<CEILDIV(NN, 8), 256, 0, stream>>>(x2, enc_g2, enc_be2, NN, HID, 1);

  hipMemsetAsync(hzero, 0, NH * sizeof(float), stream);

  // ---- TGCN stack ----
  float* outbuf[2][2] = {{oA0, oA1}, {oB0, oB1}};
  const float* in_t[2] = {x2, x2};  // layer-0 sequence is x repeated
  for (int l = 0; l < NL; ++l) {
    const float* hprev = hzero;
    for (int t = 0; t < 2; ++t) {
      // layer 0: s (and gi) are identical for both timesteps -> compute once
      bool recompute_s = !(l == 0 && t == 1);
      if (recompute_s) {
        gemm(in_t[t], gcn_W + (size_t)l * HID * HID, nullptr, xl, NN, HID, HID, false, stream);
        hipMemsetAsync(agg, 0, NH * sizeof(float), stream);
        k_edge_scatter<<<CEILDIV(NE, 8), 256, 0, stream>>>(src, dst, ew, dinv, xl, agg, NE);
        k_gcn_fin<<<CEILDIV((int)NH, 256), 256, 0, stream>>>(agg, xl, dinv, gcn_b + l * HID, (int)NH);
        gemm(agg, gru_Wih + (size_t)l * 3 * HID * HID, gru_bih + l * 3 * HID,
             gi, NN, 3 * HID, HID, true, stream);
      }
      gemm(hprev, gru_Whh + (size_t)l * 3 * HID * HID, gru_bhh + l * 3 * HID,
           gh, NN, 3 * HID, HID, true, stream);
      float* ho = outbuf[l][t];
      k_gru<<<CEILDIV((int)NH, 256), 256, 0, stream>>>(gi, gh, hprev, ho, (int)NH);
      k_ln<<<CEILDIV(NN, 8), 256, 0, stream>>>(ho, tg_g + l * HID, tg_b + l * HID, NN, HID, 0);
      hprev = ho;
    }
    in_t[0] = outbuf[l][0];
    in_t[1] = outbuf[l][1];
  }
  const float* ctx = outbuf[NL - 1][1];

  // ---- prediction head ----
  k_gather<<<CEILDIV(BATCH * ZDIM, 256), 256, 0, stream>>>(ctx, nidx, eth_tab, eth_id, zb);
  gemm(zb, p_W1, p_b1, h1, BATCH, HID, ZDIM, false, stream);
  k_ln<<<CEILDIV(BATCH, 8), 256, 0, stream>>>(h1, p_g1, p_be1, BATCH, HID, 1);
  gemm(h1, p_W2, p_b2, h2, BATCH, HID / 2, HID, false, stream);
  k_ln<<<CEILDIV(BATCH, 8), 256, 0, stream>>>(h2, p_g2, p_be2, BATCH, HID / 2, 1);
  k_head<<<CEILDIV(BATCH, 8), 256, 0, stream>>>(h2, p_W3, p_b3, out);
}